// MessagePassing_188978561156
// MI455X (gfx1250) — compile-verified
//
#include <hip/hip_runtime.h>

typedef __attribute__((ext_vector_type(2))) float v2f;
typedef __attribute__((ext_vector_type(8))) float v8f;

#define D 64

// ---------------------------------------------------------------------------
// Kernel 1: degree count via f32 atomics (deg values are small integers, exact
// in f32). deg[] must be pre-zeroed.
// ---------------------------------------------------------------------------
__global__ void deg_count_kernel(const long long* __restrict__ ei,
                                 float* __restrict__ deg, int E) {
    int e = blockIdx.x * blockDim.x + threadIdx.x;
    if (e >= E) return;
    int row = (int)ei[e];  // edge_index[0][e]
    atomicAdd(&deg[row], 1.0f);
}

// ---------------------------------------------------------------------------
// Kernel 2: deg_inv = deg > 0 ? rsqrt(deg) : 0   (in place)
// ---------------------------------------------------------------------------
__global__ void deg_inv_kernel(float* __restrict__ deg, int n) {
    int i = blockIdx.x * blockDim.x + threadIdx.x;
    if (i >= n) return;
    float d = deg[i];
    deg[i] = (d > 0.0f) ? rsqrtf(d) : 0.0f;
}

// ---------------------------------------------------------------------------
// Kernel 3: agg[row] += deg_inv[row]*deg_inv[col] * x[col]
// One thread per (edge, feature). 64 consecutive threads share an edge, so
// the gather of x[col*64 + 0..63] is a fully coalesced 256B read (L2-resident:
// x is 25.6 MB << 192 MB L2). Scatter via global_atomic_add_f32 (no return).
// agg[] must be pre-zeroed.
// ---------------------------------------------------------------------------
__global__ void edge_agg_kernel(const float* __restrict__ x,
                                const long long* __restrict__ ei,
                                const float* __restrict__ deg_inv,
                                float* __restrict__ agg, int E) {
    int t = blockIdx.x * blockDim.x + threadIdx.x;
    int e = t >> 6;        // edge id
    int d = t & (D - 1);   // feature id
    if (e >= E) return;
    int row = (int)ei[e];
    int col = (int)ei[E + e];
    float nrm = deg_inv[row] * deg_inv[col];
    atomicAdd(&agg[(size_t)row * D + d], nrm * x[(size_t)col * D + d]);
}

// ---------------------------------------------------------------------------
// Kernel 4: out = agg @ W  via V_WMMA_F32_16X16X4_F32 (full fp32 precision).
// Each wave owns one 16-row M-tile and the entire N=64 (4 accumulators).
// K=64 -> 16 k-steps of 4; 64 wmma per wave.
//
// Fragment layouts (ISA 7.12.2, wave32):
//   A (16x4 f32, 2 VGPRs): lane L, reg j -> A[L%16][2*(L/16) + j]
//   B (4x16 f32, 2 VGPRs): lane L, reg v -> B[v + 2*(L/16)][L%16]
//   C/D (16x16 f32, 8 VGPRs): lane L, reg v -> row v + 8*(L/16), col L%16
// ---------------------------------------------------------------------------
__global__ void gemm_wmma_f32_kernel(const float* __restrict__ agg,
                                     const float* __restrict__ Wm,
                                     float* __restrict__ out, int n_tiles) {
    int wave = threadIdx.x >> 5;
    int lane = threadIdx.x & 31;
    int tile = blockIdx.x * 8 + wave;
    if (tile >= n_tiles) return;   // uniform per wave -> EXEC all-ones for WMMA

    int hi = lane >> 4;   // half-wave: 0 or 1
    int lm = lane & 15;
    int m0 = tile * 16;

    const float* arow = agg + (size_t)(m0 + lm) * D + 2 * hi;
    v8f acc0 = {}, acc1 = {}, acc2 = {}, acc3 = {};

#pragma unroll
    for (int kb = 0; kb < 16; ++kb) {
        int k = kb * 4;
        // A fragment: float2 at (row m0+lm, col k + 2*hi) -- 8B aligned
        v2f a = *(const v2f*)(arow + k);

        // B fragments for the 4 N-tiles: reg v -> W[k + 2*hi + v][nt*16 + lm]
        const float* wp = Wm + (size_t)(k + 2 * hi) * D + lm;
        v2f b0, b1, b2, b3;
        b0.x = wp[0];   b0.y = wp[D];        // nt = 0
        b1.x = wp[16];  b1.y = wp[D + 16];   // nt = 1
        b2.x = wp[32];  b2.y = wp[D + 32];   // nt = 2
        b3.x = wp[48];  b3.y = wp[D + 48];   // nt = 3

        acc0 = __builtin_amdgcn_wmma_f32_16x16x4_f32(false, a, false, b0,
                                                     (short)0, acc0, false, false);
        acc1 = __builtin_amdgcn_wmma_f32_16x16x4_f32(false, a, false, b1,
                                                     (short)0, acc1, false, false);
        acc2 = __builtin_amdgcn_wmma_f32_16x16x4_f32(false, a, false, b2,
                                                     (short)0, acc2, false, false);
        acc3 = __builtin_amdgcn_wmma_f32_16x16x4_f32(false, a, false, b3,
                                                     (short)0, acc3, false, false);
    }

    // Store: reg v, lane -> out[m0 + v + 8*hi][nt*16 + lm]
    float* orow = out + (size_t)m0 * D + lm;
#pragma unroll
    for (int v = 0; v < 8; ++v) {
        size_t m = (size_t)(v + 8 * hi) * D;
        orow[m + 0]  = acc0[v];
        orow[m + 16] = acc1[v];
        orow[m + 32] = acc2[v];
        orow[m + 48] = acc3[v];
    }
}

// ---------------------------------------------------------------------------
// inputs (setup_inputs order): x[N,64] f32, r[200,64] f32 (unused),
// W[64,64] f32, edge_index[2,E] i64, edge_type[E] i64 (unused).
// output: out[N,64] f32.
// ---------------------------------------------------------------------------
extern "C" void kernel_launch(void* const* d_in, const int* in_sizes, int n_in,
                              void* d_out, int out_size, void* d_ws, size_t ws_size,
                              hipStream_t stream) {
    const float*     x  = (const float*)d_in[0];
    const float*     Wm = (const float*)d_in[2];
    const long long* ei = (const long long*)d_in[3];

    const int n_ent = in_sizes[0] / D;
    const int E     = in_sizes[4];
    float* out = (float*)d_out;

    // Workspace layout: deg/deg_inv [n_ent f32] | agg [n_ent*64 f32]
    float* deg = (float*)d_ws;
    size_t agg_off = (((size_t)n_ent * sizeof(float)) + 255) & ~(size_t)255;
    float* agg = (float*)((char*)d_ws + agg_off);

    hipMemsetAsync(deg, 0, (size_t)n_ent * sizeof(float), stream);
    hipMemsetAsync(agg, 0, (size_t)n_ent * D * sizeof(float), stream);

    deg_count_kernel<<<(E + 255) / 256, 256, 0, stream>>>(ei, deg, E);
    deg_inv_kernel<<<(n_ent + 255) / 256, 256, 0, stream>>>(deg, n_ent);

    // one thread per (edge, feature): 4 edges per 256-thread block
    edge_agg_kernel<<<(E + 3) / 4, 256, 0, stream>>>(x, ei, deg, agg, E);

    const int n_tiles = (n_ent + 15) / 16;             // 6250 (N divisible by 16)
    gemm_wmma_f32_kernel<<<(n_tiles + 7) / 8, 256, 0, stream>>>(agg, Wm, out, n_tiles);
}